// MolecularPotential_15479062134801
// MI455X (gfx1250) — compile-verified
//
#include <hip/hip_runtime.h>

// LJ pairwise potential, N=16384 atoms.
// d2 tile (16x16) computed by V_WMMA_F32_16X16X4_F32 via the K=4 embedding:
//   A[i,:] = (-2x_i, -2y_i, -2z_i, 1), B[:,j] = (x_j, y_j, z_j, |p_j|^2),
//   C[i,j] = |p_i|^2  =>  D = A*B + C = d^2(i,j).
// Epilogue: inv6 = rcp(d2)^3, lj = inv6^2 - inv6, masked by d2 < 25.
// Diagonal handling mirrors the reference (+1e6 on the diagonal): the i==j
// element can only occur in ONE wave-uniform column tile (t_diag, scalar via
// readfirstlane). On that tile a tiny side block adds a precomputed per-lane
// bias to d2 at the diagonal (r,lane) slots, pushing it past the cutoff, then
// falls into the SINGLE shared epilogue -> no duplicated epilogue, no
// accumulator merge moves, EXEC stays all-ones throughout.

typedef float v2f __attribute__((ext_vector_type(2)));
typedef float v8f __attribute__((ext_vector_type(8)));

#define N_ATOMS          16384
#define CSPLIT           4
#define CHUNK            (N_ATOMS / CSPLIT)          // 4096 columns per block
#define WAVES_PER_BLOCK  8
#define BLOCK_THREADS    (WAVES_PER_BLOCK * 32)      // 256 (8 wave32s)
#define ROWS_PER_BLOCK   (WAVES_PER_BLOCK * 16)      // 128 rows per block
#define CUTOFF2          25.0f                       // (5.0)^2
#define DIAG_BIAS        1.0e6f

__global__ __launch_bounds__(BLOCK_THREADS)
void MolecularPotential_tiles(const float* __restrict__ pos,
                              float* __restrict__ part) {
    __shared__ float4 smem[CHUNK];                   // 64 KB of the 320 KB WGP LDS

    const int tid        = threadIdx.x;
    const int chunk      = blockIdx.y;
    const int chunk_base = chunk * CHUNK;

    // Stage the column chunk: (x, y, z, |p|^2) per atom j.
    for (int idx = tid; idx < CHUNK; idx += BLOCK_THREADS) {
        const int j = chunk_base + idx;
        const float x = pos[3 * j + 0];
        const float y = pos[3 * j + 1];
        const float z = pos[3 * j + 2];
        smem[idx] = make_float4(x, y, z, x * x + y * y + z * z);
    }
    __syncthreads();

    const int wave    = tid >> 5;
    const int lane    = tid & 31;
    const int m       = lane & 15;    // row/col index within tile
    const int h       = lane >> 4;    // lane-group: selects K pair / M half
    const int rowbase = blockIdx.x * ROWS_PER_BLOCK + wave * 16;

    // --- A operand (loaded once per wave): lane L holds row m, K = {2h, 2h+1}
    const int im = rowbase + m;
    const float ax = pos[3 * im + 0];
    const float ay = pos[3 * im + 1];
    const float az = pos[3 * im + 2];
    const float sqm = ax * ax + ay * ay + az * az;
    v2f a;
    a.x = h ? (-2.0f * az) : (-2.0f * ax);
    a.y = h ? 1.0f         : (-2.0f * ay);

    // --- C operand: C vgpr r holds |p_i|^2 for row i = rowbase + r + 8h.
    // Row m's |p|^2 lives in lane m (computed above) -> gather via bpermute.
    v8f c;
#pragma unroll
    for (int r = 0; r < 8; ++r)
        c[r] = __shfl(sqm, r + 8 * h, 32);

    // Per-lane diagonal bias: in the diagonal tile, element i==j sits at
    // lane m == r + 8h for C/D vgpr r. Loop-invariant.
    v8f dbias;
#pragma unroll
    for (int r = 0; r < 8; ++r)
        dbias[r] = (m == r + 8 * h) ? DIAG_BIAS : 0.0f;

    float acc[8] = {0.f, 0.f, 0.f, 0.f, 0.f, 0.f, 0.f, 0.f};

    // The one tile index (if any) whose columns hit this wave's rows.
    // Wave-uniform; readfirstlane pins it to an SGPR -> scalar branch below.
    const int t_diag =
        __builtin_amdgcn_readfirstlane((rowbase - chunk_base) >> 4);

    const int ntiles = CHUNK / 16;                   // 256 WMMA tiles per wave
    for (int t = 0; t < ntiles; ++t) {
        const int jb = t * 16;
        // B operand from LDS: lane L holds col m, K = {2h, 2h+1}
        const float4 w = smem[jb + m];               // ds_load_b128, conflict-free
        v2f b;
        b.x = h ? w.z : w.x;
        b.y = h ? w.w : w.y;

        // d2 tile = A*B + C  (EXEC is all-ones on every path reaching here)
        v8f d = __builtin_amdgcn_wmma_f32_16x16x4_f32(
            /*neg_a=*/false, a, /*neg_b=*/false, b,
            /*c_mod=*/(short)0, c, /*reuse_a=*/false, /*reuse_b=*/false);

        // Rare, scalar-predicated side block (runs <= once per wave):
        // push the diagonal d2 past the cutoff, like the reference's +1e6.
        if (__builtin_expect(t == t_diag, 0)) {
#pragma unroll
            for (int r = 0; r < 8; ++r)
                d[r] += dbias[r];
        }

        // Single shared epilogue (cutoff mask only).
#pragma unroll
        for (int r = 0; r < 8; ++r) {
            const float d2   = fmaxf(d[r], 0.0f);
            const float inv2 = __builtin_amdgcn_rcpf(d2);         // 1/d^2
            const float inv6 = inv2 * inv2 * inv2;                // 1/d^6
            const float lj   = __builtin_fmaf(inv6, inv6, -inv6); // d^-12 - d^-6
            acc[r] += (d2 < CUTOFF2) ? lj : 0.0f;
        }
    }

    // Row sums: reduce across the 16 lanes that share h (xor masks stay in-group).
#pragma unroll
    for (int r = 0; r < 8; ++r) {
        float v = acc[r];
        v += __shfl_xor(v, 1, 32);
        v += __shfl_xor(v, 2, 32);
        v += __shfl_xor(v, 4, 32);
        v += __shfl_xor(v, 8, 32);
        acc[r] = v;
    }
    if (m == 0) {
#pragma unroll
        for (int r = 0; r < 8; ++r)
            part[chunk * N_ATOMS + rowbase + 8 * h + r] = acc[r];
    }
}

// Deterministic fixed-order reduction over the CSPLIT column chunks.
__global__ void MolecularPotential_reduce(const float* __restrict__ part,
                                          float* __restrict__ out) {
    const int i = blockIdx.x * blockDim.x + threadIdx.x;
    if (i < N_ATOMS) {
        float s = 0.0f;
#pragma unroll
        for (int cIdx = 0; cIdx < CSPLIT; ++cIdx)
            s += part[cIdx * N_ATOMS + i];
        out[i] = s;
    }
}

extern "C" void kernel_launch(void* const* d_in, const int* in_sizes, int n_in,
                              void* d_out, int out_size, void* d_ws, size_t ws_size,
                              hipStream_t stream) {
    (void)in_sizes; (void)n_in; (void)out_size; (void)ws_size;
    const float* pos  = (const float*)d_in[0];   // [16384, 3] f32
    float*       out  = (float*)d_out;           // [16384] f32
    float*       part = (float*)d_ws;            // CSPLIT * N floats = 256 KB scratch

    dim3 grid(N_ATOMS / ROWS_PER_BLOCK, CSPLIT); // (128, 4)
    MolecularPotential_tiles<<<grid, BLOCK_THREADS, 0, stream>>>(pos, part);
    MolecularPotential_reduce<<<(N_ATOMS + 255) / 256, 256, 0, stream>>>(part, out);
}